// ConvBNReLU_16088947491314
// MI455X (gfx1250) — compile-verified
//
#include <hip/hip_runtime.h>

#define N_SITES 200000
#define M_PAIRS 100000
#define K_VOL   27
#define CIN     32
#define COUT    32
#define EPSV    1e-5f

typedef float v2f __attribute__((ext_vector_type(2)));
typedef float v8f __attribute__((ext_vector_type(8)));

constexpr int TILES_PER_WAVE = 10;                        // 16 pairs each
constexpr int WAVES_PER_K    = M_PAIRS / (16 * TILES_PER_WAVE); // 625
constexpr int TOTAL_WAVES    = K_VOL * WAVES_PER_K;       // 16875

// ---------------------------------------------------------------------------
// Zero the accumulator + stats region of the workspace (graph-safe memset).
// ---------------------------------------------------------------------------
__global__ void zero_ws_kernel(float* __restrict__ p, int n) {
    int i = blockIdx.x * blockDim.x + threadIdx.x;
    int stride = gridDim.x * blockDim.x;
    for (; i < n; i += stride) p[i] = 0.0f;
}

// ---------------------------------------------------------------------------
// Sparse conv: gather 16 rows -> 16x32 * 32x32 GEMM via v_wmma_f32_16x16x4_f32
// -> scatter-add (global_atomic_add_f32) into the [N, COUT] accumulator.
// One wave per group of TILES_PER_WAVE tiles of one kernel offset k.
// ---------------------------------------------------------------------------
__global__ void conv_wmma_kernel(const float* __restrict__ x,
                                 const float* __restrict__ W,
                                 const int*   __restrict__ in_idx,
                                 const int*   __restrict__ out_idx,
                                 float*       __restrict__ acc) {
    const int lane   = threadIdx.x & 31;
    const int wid    = threadIdx.x >> 5;
    const int gwave  = blockIdx.x * (blockDim.x >> 5) + wid;
    if (gwave >= TOTAL_WAVES) return;          // whole-wave exit: EXEC stays full

    const int k      = gwave / WAVES_PER_K;
    const int wk     = gwave % WAVES_PER_K;
    const int mbase0 = wk * (16 * TILES_PER_WAVE);

    const int lanehi = lane >> 4;              // 0 | 1
    const int lanelo = lane & 15;

    // ---- Load B = W[k] (32x32), kept in registers for all tiles ----------
    // ISA 16x16x4 B layout: VGPR j, lanes[0:15] -> K = 4*kb + 2*lanehi + j,
    // N = lanelo (column half selects +16).
    const float* Wk = W + (size_t)k * CIN * COUT;
    v2f b0[8], b1[8];
#pragma unroll
    for (int kb = 0; kb < 8; ++kb) {
        const int krow = kb * 4 + lanehi * 2;
        b0[kb].x = Wk[(size_t)(krow    ) * COUT + lanelo];
        b0[kb].y = Wk[(size_t)(krow + 1) * COUT + lanelo];
        b1[kb].x = Wk[(size_t)(krow    ) * COUT + 16 + lanelo];
        b1[kb].y = Wk[(size_t)(krow + 1) * COUT + 16 + lanelo];
    }

    const int* in_k  = in_idx  + (size_t)k * M_PAIRS;
    const int* out_k = out_idx + (size_t)k * M_PAIRS;

    for (int t = 0; t < TILES_PER_WAVE; ++t) {
        const int m = mbase0 + t * 16;

        // A-row gather pointer: lane holds row M=lanelo, K-pair at 2*lanehi.
        const int arow = in_k[m + lanelo];
        const float* xrow = x + (size_t)arow * CIN + lanehi * 2;

        v8f c0 = {};
        v8f c1 = {};
#pragma unroll
        for (int kb = 0; kb < 8; ++kb) {
            v2f a = *(const v2f*)(xrow + kb * 4);   // 8B contiguous per lane
            c0 = __builtin_amdgcn_wmma_f32_16x16x4_f32(
                    false, a, false, b0[kb], (short)0, c0, false, false);
            c1 = __builtin_amdgcn_wmma_f32_16x16x4_f32(
                    false, a, false, b1[kb], (short)0, c1, false, false);
        }

        // C/D layout: VGPR i -> row m + i + 8*lanehi, col lanelo (+16).
        const int4 o0 = *(const int4*)(out_k + m + 8 * lanehi);
        const int4 o1 = *(const int4*)(out_k + m + 8 * lanehi + 4);
        const int rows[8] = {o0.x, o0.y, o0.z, o0.w, o1.x, o1.y, o1.z, o1.w};
#pragma unroll
        for (int i = 0; i < 8; ++i) {
            float* p = acc + (size_t)rows[i] * COUT + lanelo;
            atomicAdd(p,      c0[i]);
            atomicAdd(p + 16, c1[i]);
        }
    }
}

// ---------------------------------------------------------------------------
// Per-channel sum / sum-of-squares over all N sites (for batch-norm stats).
// ---------------------------------------------------------------------------
__global__ void bn_stats_kernel(const float* __restrict__ acc,
                                float* __restrict__ stats) {
    __shared__ float s_sum[8][COUT];
    __shared__ float s_sq [8][COUT];
    const int c  = threadIdx.x & 31;
    const int rg = threadIdx.x >> 5;   // 0..7 row-groups per block

    float sum = 0.0f, sq = 0.0f;
    for (int r = blockIdx.x * 8 + rg; r < N_SITES; r += gridDim.x * 8) {
        float v = acc[(size_t)r * COUT + c];
        sum += v;
        sq  += v * v;
    }
    s_sum[rg][c] = sum;
    s_sq [rg][c] = sq;
    __syncthreads();
    if (threadIdx.x < 32) {
        float a = 0.0f, b = 0.0f;
#pragma unroll
        for (int i = 0; i < 8; ++i) { a += s_sum[i][c]; b += s_sq[i][c]; }
        atomicAdd(&stats[c],        a);
        atomicAdd(&stats[COUT + c], b);
    }
}

// ---------------------------------------------------------------------------
// y = relu(gamma * (acc - mean) * rsqrt(var + eps) + beta)
// ---------------------------------------------------------------------------
__global__ void bn_relu_kernel(const float* __restrict__ acc,
                               const float* __restrict__ stats,
                               const float* __restrict__ gamma,
                               const float* __restrict__ beta,
                               float* __restrict__ y, int total) {
    const int i = blockIdx.x * blockDim.x + threadIdx.x;
    if (i >= total) return;
    const int c = i & 31;
    const float inv_n = 1.0f / (float)N_SITES;
    const float mean  = stats[c] * inv_n;
    const float var   = stats[COUT + c] * inv_n - mean * mean;
    const float v = gamma[c] * (acc[i] - mean) * rsqrtf(var + EPSV) + beta[c];
    y[i] = v > 0.0f ? v : 0.0f;
}

// ---------------------------------------------------------------------------
extern "C" void kernel_launch(void* const* d_in, const int* in_sizes, int n_in,
                              void* d_out, int out_size, void* d_ws, size_t ws_size,
                              hipStream_t stream) {
    const float* x       = (const float*)d_in[0];
    const float* W       = (const float*)d_in[1];
    const float* gamma   = (const float*)d_in[2];
    const float* beta    = (const float*)d_in[3];
    const int*   in_idx  = (const int*)d_in[4];
    const int*   out_idx = (const int*)d_in[5];
    float*       y       = (float*)d_out;

    float* acc   = (float*)d_ws;                       // [N, COUT] fp32
    float* stats = acc + (size_t)N_SITES * COUT;       // [2, COUT] fp32

    const int zero_n = N_SITES * COUT + 2 * COUT;
    zero_ws_kernel<<<1024, 256, 0, stream>>>(acc, zero_n);

    const int waves_per_block = 8;                     // 256 threads
    const int blocks = (TOTAL_WAVES + waves_per_block - 1) / waves_per_block;
    conv_wmma_kernel<<<blocks, 256, 0, stream>>>(x, W, in_idx, out_idx, acc);

    bn_stats_kernel<<<512, 256, 0, stream>>>(acc, stats);

    const int total = N_SITES * COUT;
    bn_relu_kernel<<<(total + 255) / 256, 256, 0, stream>>>(acc, stats, gamma,
                                                            beta, y, total);
}